// LSTMModel_13666585936110
// MI455X (gfx1250) — compile-verified
//
#include <hip/hip_runtime.h>
#include <hip/hip_bf16.h>

typedef __attribute__((ext_vector_type(2))) float v2f;
typedef __attribute__((ext_vector_type(4))) float v4f;
typedef __attribute__((ext_vector_type(8))) float v8f;

#define BATCH  256
#define SEQ    256
#define INSZ   128
#define HIDDEN 512

// Fused LSTM timestep:
//   gates[b, q*H+n] = sum_k [x_t | h_prev][b,k] * [w_ih | w_hh][q*H+n, k]  (+bias at pointwise)
//   then i/f/g/o activations, c/h update.
// Block = 128 threads (4 waves) -> 32x16 (batch x hidden) tile, all 4 gates
// (two 16x16 M-tiles per wave share the B operands => 8 WMMA per 6 loads).
// K reduction split across the 4 waves in contiguous scalar-bound slices; each slice
// intersects x-region [0,K1) and h-region [K1,K1+H) contiguously -> two branch-free loops.
template <int K1, int XSTRIDE>
__global__ __launch_bounds__(128, 1) void lstm_step_kernel(
    const float* __restrict__ xin,   // A source for k < K1 ; row stride = XSTRIDE
    const float* __restrict__ w_ih,  // [4H, K1]
    const float* __restrict__ w_hh,  // [4H, H]
    const float* __restrict__ b_ih,  // [4H]
    const float* __restrict__ b_hh,  // [4H]
    const float* __restrict__ h_prev,// [B, H] or unused when has_prev==0
    float* __restrict__ cbuf,        // [B, H] in/out (ignored on read when has_prev==0)
    float* __restrict__ h_out,       // [B, H]
    int has_prev)
{
    const int lane  = threadIdx.x & 31;
    // Scalarize the wave id so K-loop bounds live in SGPRs (scalar loop control).
    const int wave  = __builtin_amdgcn_readfirstlane(threadIdx.x >> 5);
    const int b0    = blockIdx.x << 5;   // batch tile base (32 rows)
    const int h0    = blockIdx.y << 4;   // hidden tile base (16 cols)
    const int m16   = lane & 15;         // A row / B col within 16-tile
    const int khalf = lane >> 4;         // selects K pair {0,1} vs {2,3}

    constexpr int Ktot   = K1 + HIDDEN;
    constexpr int kslice = Ktot >> 2;    // per-wave K slice (160 or 256; multiple of 4)
    const int kbeg = wave * kslice;
    const int kend = kbeg + kslice;

    // Intersect this wave's slice with the two A-source regions.
    const int xbeg = kbeg < K1 ? kbeg : K1;
    const int xend = kend < K1 ? kend : K1;
    const int hbeg = (kbeg > K1 ? kbeg : K1) - K1;
    const int hend = (kend > K1 ? kend : K1) - K1;
    const int hend_eff = has_prev ? hend : hbeg;   // zero-trip at t==0 (no CFG join)

    const int nGlob = h0 + m16;          // hidden column this lane supplies B for

    v8f accLo[4], accHi[4];
    accLo[0] = (v8f){0.f,0.f,0.f,0.f,0.f,0.f,0.f,0.f};
    accLo[1] = accLo[0]; accLo[2] = accLo[0]; accLo[3] = accLo[0];
    accHi[0] = accLo[0]; accHi[1] = accLo[0]; accHi[2] = accLo[0]; accHi[3] = accLo[0];

    const float* aXlo = xin + (size_t)(b0 + m16) * XSTRIDE + (khalf << 1);
    const float* aXhi = xin + (size_t)(b0 + 16 + m16) * XSTRIDE + (khalf << 1);
    const float* aHlo = h_prev + (size_t)(b0 + m16) * HIDDEN + (khalf << 1);
    const float* aHhi = h_prev + (size_t)(b0 + 16 + m16) * HIDDEN + (khalf << 1);

    const float* wi0 = w_ih + (size_t)nGlob * K1 + (khalf << 1);
    const float* wi1 = wi0 + (size_t)HIDDEN * K1;
    const float* wi2 = wi1 + (size_t)HIDDEN * K1;
    const float* wi3 = wi2 + (size_t)HIDDEN * K1;
    const float* wh0 = w_hh + (size_t)nGlob * HIDDEN + (khalf << 1);
    const float* wh1 = wh0 + (size_t)HIDDEN * HIDDEN;
    const float* wh2 = wh1 + (size_t)HIDDEN * HIDDEN;
    const float* wh3 = wh2 + (size_t)HIDDEN * HIDDEN;

    // ---- x-region ----
    #pragma unroll 4
    for (int k4 = xbeg; k4 < xend; k4 += 4) {
        const v2f aLo = *(const v2f*)(aXlo + k4);
        const v2f aHi = *(const v2f*)(aXhi + k4);
        const v2f bq0 = *(const v2f*)(wi0 + k4);
        const v2f bq1 = *(const v2f*)(wi1 + k4);
        const v2f bq2 = *(const v2f*)(wi2 + k4);
        const v2f bq3 = *(const v2f*)(wi3 + k4);
        accLo[0] = __builtin_amdgcn_wmma_f32_16x16x4_f32(false, aLo, false, bq0, (short)0, accLo[0], false, false);
        accHi[0] = __builtin_amdgcn_wmma_f32_16x16x4_f32(false, aHi, false, bq0, (short)0, accHi[0], false, false);
        accLo[1] = __builtin_amdgcn_wmma_f32_16x16x4_f32(false, aLo, false, bq1, (short)0, accLo[1], false, false);
        accHi[1] = __builtin_amdgcn_wmma_f32_16x16x4_f32(false, aHi, false, bq1, (short)0, accHi[1], false, false);
        accLo[2] = __builtin_amdgcn_wmma_f32_16x16x4_f32(false, aLo, false, bq2, (short)0, accLo[2], false, false);
        accHi[2] = __builtin_amdgcn_wmma_f32_16x16x4_f32(false, aHi, false, bq2, (short)0, accHi[2], false, false);
        accLo[3] = __builtin_amdgcn_wmma_f32_16x16x4_f32(false, aLo, false, bq3, (short)0, accLo[3], false, false);
        accHi[3] = __builtin_amdgcn_wmma_f32_16x16x4_f32(false, aHi, false, bq3, (short)0, accHi[3], false, false);
    }

    // ---- h-region (zero trips at t==0) ----
    #pragma unroll 4
    for (int k4 = hbeg; k4 < hend_eff; k4 += 4) {
        const v2f aLo = *(const v2f*)(aHlo + k4);
        const v2f aHi = *(const v2f*)(aHhi + k4);
        const v2f bq0 = *(const v2f*)(wh0 + k4);
        const v2f bq1 = *(const v2f*)(wh1 + k4);
        const v2f bq2 = *(const v2f*)(wh2 + k4);
        const v2f bq3 = *(const v2f*)(wh3 + k4);
        accLo[0] = __builtin_amdgcn_wmma_f32_16x16x4_f32(false, aLo, false, bq0, (short)0, accLo[0], false, false);
        accHi[0] = __builtin_amdgcn_wmma_f32_16x16x4_f32(false, aHi, false, bq0, (short)0, accHi[0], false, false);
        accLo[1] = __builtin_amdgcn_wmma_f32_16x16x4_f32(false, aLo, false, bq1, (short)0, accLo[1], false, false);
        accHi[1] = __builtin_amdgcn_wmma_f32_16x16x4_f32(false, aHi, false, bq1, (short)0, accHi[1], false, false);
        accLo[2] = __builtin_amdgcn_wmma_f32_16x16x4_f32(false, aLo, false, bq2, (short)0, accLo[2], false, false);
        accHi[2] = __builtin_amdgcn_wmma_f32_16x16x4_f32(false, aHi, false, bq2, (short)0, accHi[2], false, false);
        accLo[3] = __builtin_amdgcn_wmma_f32_16x16x4_f32(false, aLo, false, bq3, (short)0, accLo[3], false, false);
        accHi[3] = __builtin_amdgcn_wmma_f32_16x16x4_f32(false, aHi, false, bq3, (short)0, accHi[3], false, false);
    }

    // Dump per-wave partial tiles to LDS, layout [wave][gate][n*32 + m]:
    // C/D layout: lane&15 = n ; VGPR r holds m = r + 8*(lane>>4)  -> 8 contiguous m per lane
    // => two ds_store_b128 per gate per M-tile.
    __shared__ float red[4][4][512];   // 32 KB
    #pragma unroll
    for (int q = 0; q < 4; ++q) {
        const int base = (m16 << 5) + (khalf << 3);
        *(v4f*)&red[wave][q][base]          = (v4f){accLo[q][0], accLo[q][1], accLo[q][2], accLo[q][3]};
        *(v4f*)&red[wave][q][base + 4]      = (v4f){accLo[q][4], accLo[q][5], accLo[q][6], accLo[q][7]};
        *(v4f*)&red[wave][q][base + 16]     = (v4f){accHi[q][0], accHi[q][1], accHi[q][2], accHi[q][3]};
        *(v4f*)&red[wave][q][base + 16 + 4] = (v4f){accHi[q][4], accHi[q][5], accHi[q][6], accHi[q][7]};
    }
    __syncthreads();

    // Reduce across the 4 waves + bias + LSTM cell pointwise. 512 elements / 128 threads.
    // e -> (mm = e>>4, nn = e&15): consecutive threads hit consecutive nn => coalesced
    // global writes (16-float runs per 16 threads).
    for (int e = threadIdx.x; e < 512; e += 128) {
        const int mm = e >> 4;
        const int nn = e & 15;
        const int l  = (nn << 5) + mm;   // LDS index
        const int gcol = h0 + nn;

        float gi = red[0][0][l] + red[1][0][l] + red[2][0][l] + red[3][0][l]
                 + b_ih[gcol] + b_hh[gcol];
        float gf = red[0][1][l] + red[1][1][l] + red[2][1][l] + red[3][1][l]
                 + b_ih[HIDDEN + gcol] + b_hh[HIDDEN + gcol];
        float gg = red[0][2][l] + red[1][2][l] + red[2][2][l] + red[3][2][l]
                 + b_ih[2 * HIDDEN + gcol] + b_hh[2 * HIDDEN + gcol];
        float go = red[0][3][l] + red[1][3][l] + red[2][3][l] + red[3][3][l]
                 + b_ih[3 * HIDDEN + gcol] + b_hh[3 * HIDDEN + gcol];

        const float ii = 1.f / (1.f + __expf(-gi));
        const float ff = 1.f / (1.f + __expf(-gf));
        const float gt = tanhf(gg);
        const float oo = 1.f / (1.f + __expf(-go));

        const size_t idx = (size_t)(b0 + mm) * HIDDEN + gcol;
        const float cp = has_prev ? cbuf[idx] : 0.f;
        const float cn = ff * cp + ii * gt;
        cbuf[idx]  = cn;
        h_out[idx] = oo * tanhf(cn);
    }
}

// out[b] = dot(h_last[b,:], w_fc) + b_fc
__global__ __launch_bounds__(128, 1) void fc_kernel(
    const float* __restrict__ hlast, const float* __restrict__ w_fc,
    const float* __restrict__ b_fc, float* __restrict__ out)
{
    const int b = blockIdx.x;
    float s = 0.f;
    for (int h = threadIdx.x; h < HIDDEN; h += 128)
        s += hlast[(size_t)b * HIDDEN + h] * w_fc[h];
    __shared__ float sh[128];
    sh[threadIdx.x] = s;
    __syncthreads();
    for (int off = 64; off > 0; off >>= 1) {
        if (threadIdx.x < off) sh[threadIdx.x] += sh[threadIdx.x + off];
        __syncthreads();
    }
    if (threadIdx.x == 0) out[b] = sh[0] + b_fc[0];
}

extern "C" void kernel_launch(void* const* d_in, const int* in_sizes, int n_in,
                              void* d_out, int out_size, void* d_ws, size_t ws_size,
                              hipStream_t stream) {
    (void)in_sizes; (void)n_in; (void)out_size; (void)ws_size;
    const float* x     = (const float*)d_in[0];   // [B, T, I]
    const float* w_ih0 = (const float*)d_in[1];   // [4H, I]
    const float* w_hh0 = (const float*)d_in[2];   // [4H, H]
    const float* b_ih0 = (const float*)d_in[3];
    const float* b_hh0 = (const float*)d_in[4];
    const float* w_ih1 = (const float*)d_in[5];   // [4H, H]
    const float* w_hh1 = (const float*)d_in[6];   // [4H, H]
    const float* b_ih1 = (const float*)d_in[7];
    const float* b_hh1 = (const float*)d_in[8];
    const float* w_fc  = (const float*)d_in[9];   // [1, H]
    const float* b_fc  = (const float*)d_in[10];  // [1]
    float* out = (float*)d_out;

    float* ws    = (float*)d_ws;
    float* hs1   = ws;                                     // [T, B, H]  layer-0 hidden history
    float* hping = hs1 + (size_t)SEQ * BATCH * HIDDEN;     // [2, B, H]  layer-1 h ping-pong
    float* cbuf  = hping + (size_t)2 * BATCH * HIDDEN;     // [B, H]     cell state (reused per layer)

    const dim3 grid(BATCH / 32, HIDDEN / 16);
    const dim3 blk(128);

    // Layer 0: A = [x_t | h_{t-1}], x row stride = T*I (x is [B,T,I])
    for (int t = 0; t < SEQ; ++t) {
        const float* xt = x + (size_t)t * INSZ;
        const float* hp = (t == 0) ? hs1 : hs1 + (size_t)(t - 1) * BATCH * HIDDEN;
        float* ho = hs1 + (size_t)t * BATCH * HIDDEN;
        lstm_step_kernel<INSZ, SEQ * INSZ><<<grid, blk, 0, stream>>>(
            xt, w_ih0, w_hh0, b_ih0, b_hh0, hp, cbuf, ho, (t != 0) ? 1 : 0);
    }

    // Layer 1: A = [hs1[t] | h_{t-1}], hs1[t] is contiguous [B,H]
    for (int t = 0; t < SEQ; ++t) {
        const float* xt = hs1 + (size_t)t * BATCH * HIDDEN;
        const float* hp = hping + (size_t)((t + 1) & 1) * BATCH * HIDDEN;
        float* ho = hping + (size_t)(t & 1) * BATCH * HIDDEN;
        lstm_step_kernel<HIDDEN, HIDDEN><<<grid, blk, 0, stream>>>(
            xt, w_ih1, w_hh1, b_ih1, b_hh1, hp, cbuf, ho, (t != 0) ? 1 : 0);
    }

    // Final FC on h at t = SEQ-1 (parity (SEQ-1)&1 == 1)
    const float* hlast = hping + (size_t)((SEQ - 1) & 1) * BATCH * HIDDEN;
    fc_kernel<<<dim3(BATCH), blk, 0, stream>>>(hlast, w_fc, b_fc, out);
}